// GraphAttentionBlock_20366734917865
// MI455X (gfx1250) — compile-verified
//
#include <hip/hip_runtime.h>
#include <hip/hip_bf16.h>
#include <math.h>

#define DHID 128
#define CPH 64
#define NEG_SLOPE 0.2f
#define BN_EPS 1e-5f

typedef float v2f __attribute__((ext_vector_type(2)));
typedef float v8f __attribute__((ext_vector_type(8)));

// ---------------------------------------------------------------------------
// GEMM: Y[M,128] = X[M,128] * W[128,128], M a multiple of 16 (50000 = 16*3125)
// One block = 256 threads = 8 waves; block owns a 16-row strip, wave w owns
// columns [16w, 16w+16). Full-precision V_WMMA_F32_16X16X4_F32, K-loop of 32.
// A layout (16x4 f32): lane L holds M=L&15, K = 2*(L>>4) + {vgpr0,vgpr1}.
// B layout (4x16 f32): lane L holds N=L&15, K = 2*(L>>4) + {vgpr0,vgpr1}.
// C/D layout (16x16 f32): vgpr v -> M = v + 8*(L>>4), N = L&15.
// ---------------------------------------------------------------------------
__global__ __launch_bounds__(256) void gemm_wmma_f32(const float* __restrict__ X,
                                                     const float* __restrict__ Wm,
                                                     float* __restrict__ Y) {
  __shared__ float xs[16 * 128];
  const int strip = blockIdx.x;
  const float* xsrc = X + (size_t)strip * 16 * 128;
  for (int i = threadIdx.x; i < 16 * 128; i += 256) xs[i] = xsrc[i];
  __syncthreads();

  const int lane = threadIdx.x & 31;
  const int wave = threadIdx.x >> 5;          // 0..7 -> column tile
  const int m    = lane & 15;                 // A row within tile
  const int ko   = (lane >> 4) * 2;           // K pair selector
  const int col  = wave * 16 + (lane & 15);   // global output column

  v8f acc = {};
  for (int k = 0; k < 128; k += 4) {
    v2f a, b;
    a[0] = xs[m * 128 + k + ko];
    a[1] = xs[m * 128 + k + ko + 1];
    b[0] = Wm[(k + ko) * 128 + col];
    b[1] = Wm[(k + ko + 1) * 128 + col];
    acc = __builtin_amdgcn_wmma_f32_16x16x4_f32(false, a, false, b,
                                                (short)0, acc, false, false);
  }

  const int rbase = strip * 16 + ((lane >> 4) * 8);
#pragma unroll
  for (int v = 0; v < 8; ++v)
    Y[(size_t)(rbase + v) * 128 + col] = acc[v];
}

// ---------------------------------------------------------------------------
// a_src[n,h] = <h[n,h,:], att_src[h,:]>, a_dst likewise. One thread per (n,h).
// ---------------------------------------------------------------------------
__global__ void attn_coef(const float* __restrict__ h,
                          const float* __restrict__ att_s,
                          const float* __restrict__ att_d,
                          float* __restrict__ a_s, float* __restrict__ a_d,
                          int n2) {
  int t = blockIdx.x * blockDim.x + threadIdx.x;
  if (t >= n2) return;
  const int head = t & 1;
  const float* hp = h + (size_t)(t >> 1) * DHID + head * CPH;
  const float* sp = att_s + head * CPH;
  const float* dp = att_d + head * CPH;
  float s1 = 0.f, s2 = 0.f;
#pragma unroll 4
  for (int c = 0; c < CPH; ++c) {
    float v = hp[c];
    s1 += v * sp[c];
    s2 += v * dp[c];
  }
  a_s[t] = s1;
  a_d[t] = s2;
}

// ---------------------------------------------------------------------------
// Per-layer init: agg <- broadcast(bias) (folds the "+bias"), m <- -big,
// s <- 0, column stats <- 0. Runs every call (workspace is never re-poisoned).
// ---------------------------------------------------------------------------
__global__ void init_layer(float* __restrict__ agg, const float* __restrict__ bias,
                           float* __restrict__ m, float* __restrict__ s,
                           float* __restrict__ cs, float* __restrict__ csq,
                           int n128, int n2) {
  int i = blockIdx.x * blockDim.x + threadIdx.x;
  if (i < n128) agg[i] = bias[i & 127];
  if (i < n2) { m[i] = -1e30f; s[i] = 0.f; }
  if (i < 128) { cs[i] = 0.f; csq[i] = 0.f; }
}

// int-punned float atomic max (works for mixed signs given negative init)
__device__ __forceinline__ void atomicMaxF(float* addr, float val) {
  if (val >= 0.f)
    atomicMax((int*)addr, __float_as_int(val));
  else
    atomicMin((unsigned int*)addr, __float_as_uint(val));
}

// Pass 1: e = leaky_relu(a_src[src]+a_dst[dst]); segment max over dst.
__global__ void edge_lrelu_max(const long long* __restrict__ ei, int E_, int EN,
                               const float* __restrict__ a_s,
                               const float* __restrict__ a_d,
                               float* __restrict__ ebuf, float* __restrict__ m) {
  int t = blockIdx.x * blockDim.x + threadIdx.x;
  if (t >= EN * 2) return;
  const int e = t >> 1, head = t & 1;
  int src, dst;
  if (e < E_) { src = (int)ei[e]; dst = (int)ei[E_ + e]; }
  else        { src = dst = e - E_; }                       // self-loops
  float v = a_s[src * 2 + head] + a_d[dst * 2 + head];
  v = v > 0.f ? v : NEG_SLOPE * v;
  ebuf[t] = v;
  atomicMaxF(&m[dst * 2 + head], v);
}

// Pass 2: ex = exp(e - m[dst]); segment sum over dst. ebuf overwritten with ex.
__global__ void edge_exp_sum(const long long* __restrict__ ei, int E_, int EN,
                             const float* __restrict__ m,
                             float* __restrict__ ebuf, float* __restrict__ s) {
  int t = blockIdx.x * blockDim.x + threadIdx.x;
  if (t >= EN * 2) return;
  const int e = t >> 1, head = t & 1;
  const int dst = (e < E_) ? (int)ei[E_ + e] : (e - E_);
  float ex = __expf(ebuf[t] - m[dst * 2 + head]);
  ebuf[t] = ex;
  atomicAdd(&s[dst * 2 + head], ex);
}

// Pass 3: agg[dst] += alpha * h[src]. One wave per edge: float4 gather of the
// 512B h-row (L2-resident: h fits in MI455X's 192MB L2), 4 f32 atomics/lane.
// Lane l covers channels [4l,4l+4) -> head = l>>4.
__global__ __launch_bounds__(256) void edge_aggregate(
    const long long* __restrict__ ei, int E_, int EN,
    const float* __restrict__ h, const float* __restrict__ ebuf,
    const float* __restrict__ s, float* __restrict__ agg) {
  const int wv = (blockIdx.x * 256 + threadIdx.x) >> 5;
  const int lane = threadIdx.x & 31;
  if (wv >= EN) return;
  int src, dst;
  if (wv < E_) { src = (int)ei[wv]; dst = (int)ei[E_ + wv]; }
  else         { src = dst = wv - E_; }
  const int head = lane >> 4;
  const float alpha = ebuf[wv * 2 + head] / (s[dst * 2 + head] + 1e-16f);
  const float4 hv = ((const float4*)(h + (size_t)src * DHID))[lane];
  float* ap = agg + (size_t)dst * DHID + lane * 4;
  atomicAdd(ap + 0, alpha * hv.x);
  atomicAdd(ap + 1, alpha * hv.y);
  atomicAdd(ap + 2, alpha * hv.z);
  atomicAdd(ap + 3, alpha * hv.w);
}

// BN pass 1: per-column sum / sum-of-squares. 128 threads = 128 adjacent
// columns -> fully coalesced row reads; one atomicAdd pair per block/column.
__global__ __launch_bounds__(128) void bn_stats(const float* __restrict__ x, int n,
                                                float* __restrict__ cs,
                                                float* __restrict__ csq) {
  const int col = threadIdx.x;
  const int r0 = blockIdx.x * 256;
  float s = 0.f, q = 0.f;
  for (int i = 0; i < 256; ++i) {
    int r = r0 + i;
    if (r < n) {
      float v = x[(size_t)r * 128 + col];
      s += v;
      q += v * v;
    }
  }
  atomicAdd(&cs[col], s);
  atomicAdd(&csq[col], q);
}

// BN pass 2 + ELU.
__global__ void bn_elu(const float* __restrict__ x, const float* __restrict__ cs,
                       const float* __restrict__ csq,
                       const float* __restrict__ gamma,
                       const float* __restrict__ beta,
                       float* __restrict__ y, int n128, float invN) {
  int i = blockIdx.x * blockDim.x + threadIdx.x;
  if (i >= n128) return;
  const int col = i & 127;
  const float mean = cs[col] * invN;
  const float var = csq[col] * invN - mean * mean;
  float v = (x[i] - mean) * rsqrtf(var + BN_EPS) * gamma[col] + beta[col];
  y[i] = v > 0.f ? v : expm1f(v);
}

// ---------------------------------------------------------------------------
extern "C" void kernel_launch(void* const* d_in, const int* in_sizes, int n_in,
                              void* d_out, int out_size, void* d_ws, size_t ws_size,
                              hipStream_t stream) {
  const float*     x   = (const float*)d_in[0];
  const long long* ei  = (const long long*)d_in[1];
  const float*     W1  = (const float*)d_in[2];
  const float*     as1 = (const float*)d_in[3];
  const float*     ad1 = (const float*)d_in[4];
  const float*     b1  = (const float*)d_in[5];
  const float*     W2  = (const float*)d_in[6];
  const float*     as2 = (const float*)d_in[7];
  const float*     ad2 = (const float*)d_in[8];
  const float*     b2  = (const float*)d_in[9];
  const float*     g1  = (const float*)d_in[10];
  const float*     be1 = (const float*)d_in[11];
  const float*     g2  = (const float*)d_in[12];
  const float*     be2 = (const float*)d_in[13];

  const int N    = in_sizes[0] / DHID;   // 50000
  const int E    = in_sizes[1] / 2;      // 800000
  const int EN   = E + N;                // edges incl. self loops
  const int n128 = N * DHID;
  const int n2   = N * 2;

  float* ws   = (float*)d_ws;
  float* bufA = ws; ws += n128;          // h / ping
  float* bufB = ws; ws += n128;          // agg / pong
  float* a_s  = ws; ws += n2;
  float* a_d  = ws; ws += n2;
  float* mbuf = ws; ws += n2;
  float* sbuf = ws; ws += n2;
  float* ebuf = ws; ws += (size_t)EN * 2;
  float* cs   = ws; ws += 128;
  float* csq  = ws; ws += 128;

  auto layer = [&](const float* xin, const float* Wl, const float* asrc,
                   const float* adst, const float* bias, const float* gamma,
                   const float* beta, float* hbuf, float* aggbuf, float* outbuf) {
    gemm_wmma_f32<<<N / 16, 256, 0, stream>>>(xin, Wl, hbuf);
    attn_coef<<<(n2 + 255) / 256, 256, 0, stream>>>(hbuf, asrc, adst, a_s, a_d, n2);
    init_layer<<<(n128 + 255) / 256, 256, 0, stream>>>(aggbuf, bias, mbuf, sbuf,
                                                       cs, csq, n128, n2);
    edge_lrelu_max<<<(EN * 2 + 255) / 256, 256, 0, stream>>>(ei, E, EN, a_s, a_d,
                                                             ebuf, mbuf);
    edge_exp_sum<<<(EN * 2 + 255) / 256, 256, 0, stream>>>(ei, E, EN, mbuf, ebuf,
                                                           sbuf);
    edge_aggregate<<<(EN + 7) / 8, 256, 0, stream>>>(ei, E, EN, hbuf, ebuf, sbuf,
                                                     aggbuf);
    bn_stats<<<(N + 255) / 256, 128, 0, stream>>>(aggbuf, N, cs, csq);
    bn_elu<<<(n128 + 255) / 256, 256, 0, stream>>>(aggbuf, cs, csq, gamma, beta,
                                                   outbuf, n128, 1.0f / (float)N);
  };

  // Layer 1: h1 in bufA, agg in bufB, BN+ELU back into bufA (h1 dead by then).
  layer(x, W1, as1, ad1, b1, g1, be1, bufA, bufB, bufA);
  // Layer 2: input bufA, h2 in bufB, agg in bufA (input dead), final -> d_out.
  layer(bufA, W2, as2, ad2, b2, g2, be2, bufB, bufA, (float*)d_out);
}